// BoxPairMultiScaleRoIAlign_69234872811939
// MI455X (gfx1250) — compile-verified
//
#include <hip/hip_runtime.h>
#include <cstdint>
#include <cstddef>

#define OUTS 7
#define SAMP 14          // OUTS * sampling_ratio(2)
#define CCH  256

typedef int v4i __attribute__((ext_vector_type(4)));
typedef __attribute__((address_space(1))) void* as1_void_p;
typedef __attribute__((address_space(3))) void* as3_void_p;
typedef __attribute__((address_space(1))) v4i*  as1_v4i_p;
typedef __attribute__((address_space(3))) v4i*  as3_v4i_p;

// ---- probe CDNA5 async global<->LDS builtins (gfx1250) ----
#if defined(__has_builtin)
# if __has_builtin(__builtin_amdgcn_global_store_async_from_lds_b128)
#  define HAVE_ASYNC_B128 1
# endif
# if __has_builtin(__builtin_amdgcn_s_wait_asynccnt)
#  define HAVE_WAIT_ASYNC 1
# endif
#endif

// ---------------------------------------------------------------------------
// Kernel 1: NCHW -> NHWC transpose of one pyramid level (coalesced both ways)
// src: [B, C, HW]   dst: [B, HW, C]
// ---------------------------------------------------------------------------
__global__ void nchw_to_nhwc(const float* __restrict__ src,
                             float* __restrict__ dst,
                             int C, int HW) {
  __shared__ float tile[32][33];
  const int b  = blockIdx.z;
  const int p0 = blockIdx.x * 32;
  const int c0 = blockIdx.y * 32;
  const int tx = threadIdx.x, ty = threadIdx.y;
#pragma unroll
  for (int j = 0; j < 4; ++j) {
    int c = c0 + ty + j * 8;
    int p = p0 + tx;
    float v = 0.0f;
    if (p < HW && c < C) v = src[((size_t)b * C + c) * HW + p];
    tile[ty + j * 8][tx] = v;
  }
  __syncthreads();
#pragma unroll
  for (int j = 0; j < 4; ++j) {
    int p = p0 + ty + j * 8;
    int c = c0 + tx;
    if (p < HW && c < C)
      dst[((size_t)b * HW + p) * C + c] = tile[tx][ty + j * 8];
  }
}

// ---------------------------------------------------------------------------
// Kernel 2: one block (256 threads = 8 wave32) per ROI pair.
// lane == channel -> every bilinear corner gather is a fully coalesced
// 1KB (256 x f32) vector load when layout is NHWC.
// ---------------------------------------------------------------------------
__global__ __launch_bounds__(CCH) void
roi_align_pairs(const float* __restrict__ l0, const float* __restrict__ l1,
                const float* __restrict__ l2, const float* __restrict__ l3,
                const float* __restrict__ bxa, const float* __restrict__ bxb,
                const int* __restrict__ bidx, float* __restrict__ out,
                int B, int nhwc) {
  __shared__ int   s_i0[2 * SAMP];
  __shared__ int   s_i1[2 * SAMP];
  __shared__ float s_h [2 * SAMP];
  __shared__ float s_l [2 * SAMP];
  __shared__ __align__(16) float s_out[CCH * OUTS * OUTS];   // 12544 f32 = 50176 B

  const int n = blockIdx.x;
  const int t = threadIdx.x;

  // ---- union box + FPN level (uniform per block; computed redundantly) ----
  const float a0 = bxa[n * 4 + 0], a1 = bxa[n * 4 + 1];
  const float a2 = bxa[n * 4 + 2], a3 = bxa[n * 4 + 3];
  const float b0 = bxb[n * 4 + 0], b1 = bxb[n * 4 + 1];
  const float b2 = bxb[n * 4 + 2], b3 = bxb[n * 4 + 3];
  const float ux1 = fminf(a0, b0), uy1 = fminf(a1, b1);
  const float ux2 = fmaxf(a2, b2), uy2 = fmaxf(a3, b3);

  const float area = (ux2 - ux1) * (uy2 - uy1);
  float lf = floorf(4.0f + log2f(sqrtf(area) / 224.0f + 1e-6f));
  const int lvl = (int)fminf(fmaxf(lf, 2.0f), 5.0f) - 2;     // 0..3

  const int Hs[4] = {200, 100, 50, 25};
  const int H = Hs[lvl];
  const int W = H;
  const size_t HW = (size_t)H * W;
  const float scale = 1.0f / (float)(4 << lvl);              // 1/4..1/32
  const float* feat = (lvl == 0) ? l0 : (lvl == 1) ? l1 : (lvl == 2) ? l2 : l3;
  const int bb = bidx[n];

  const float x1 = ux1 * scale, y1 = uy1 * scale;
  const float rw = fmaxf(ux2 * scale - x1, 1.0f);
  const float rh = fmaxf(uy2 * scale - y1, 1.0f);

  // ---- per-axis sample metadata (validity folded into zeroed weights) ----
  if (t < 2 * SAMP) {
    const bool isY  = t < SAMP;
    const int  s    = isY ? t : t - SAMP;
    const float st  = isY ? y1 : x1;
    const float len = isY ? rh : rw;
    const float lim = (float)H;                              // square levels
    const float coord = st + ((float)s + 0.5f) * (len * (1.0f / (float)SAMP));
    const bool valid = (coord >= -1.0f) && (coord <= lim);
    const float cc = fminf(fmaxf(coord, 0.0f), lim - 1.0f);
    const int i0 = (int)floorf(cc);
    const int i1 = min(i0 + 1, H - 1);
    float l = cc - (float)i0;
    float h = 1.0f - l;
    if (!valid) { h = 0.0f; l = 0.0f; }
    s_i0[t] = i0; s_i1[t] = i1; s_h[t] = h; s_l[t] = l;
  }
  __syncthreads();

  const int c = t;                                           // lane == channel
  const size_t bbase = (size_t)bb * HW;

  for (int o = 0; o < OUTS * OUTS; ++o) {
    const int oy = o / OUTS, ox = o % OUTS;
    float acc = 0.0f;
#pragma unroll
    for (int iy = 0; iy < 2; ++iy) {
      const int sy = oy * 2 + iy;
      const int y0 = s_i0[sy], yI = s_i1[sy];
      const float hy = s_h[sy], ly = s_l[sy];
#pragma unroll
      for (int ix = 0; ix < 2; ++ix) {
        const int sx = ox * 2 + ix;
        const int x0 = s_i0[SAMP + sx], xI = s_i1[SAMP + sx];
        const float hx = s_h[SAMP + sx], lx = s_l[SAMP + sx];
        float v00, v01, v10, v11;
        if (nhwc) {
          const float* r0 = feat + (bbase + (size_t)y0 * W) * CCH;
          const float* r1 = feat + (bbase + (size_t)yI * W) * CCH;
          v00 = r0[(size_t)x0 * CCH + c];  v01 = r0[(size_t)xI * CCH + c];
          v10 = r1[(size_t)x0 * CCH + c];  v11 = r1[(size_t)xI * CCH + c];
        } else {
          const float* pc = feat + ((size_t)bb * CCH + c) * HW;
          v00 = pc[(size_t)y0 * W + x0];   v01 = pc[(size_t)y0 * W + xI];
          v10 = pc[(size_t)yI * W + x0];   v11 = pc[(size_t)yI * W + xI];
        }
        acc += hy * (hx * v00 + lx * v01) + ly * (hx * v10 + lx * v11);
      }
    }
    // stride-49 per lane: gcd(49,64)=1 -> bank-conflict-free under wave32
    s_out[c * (OUTS * OUTS) + o] = acc * 0.25f;
  }
  __syncthreads();

  // ---- coalesced async writeback: 12544 f32 = 3136 x b128 ----
  float* obase = out + (size_t)n * (CCH * OUTS * OUTS);
#ifdef HAVE_ASYNC_B128
#pragma unroll
  for (int j = 0; j < 12; ++j) {
    const int off = j * 1024 + t * 4;
    __builtin_amdgcn_global_store_async_from_lds_b128(
        (as1_v4i_p)(as1_void_p)(obase + off),
        (as3_v4i_p)(as3_void_p)(&s_out[off]), 0, 0);
  }
  if (t < 64) {   // tail 256 floats; async ops are per-lane EXEC-masked (legal)
    const int off = 12 * 1024 + t * 4;
    __builtin_amdgcn_global_store_async_from_lds_b128(
        (as1_v4i_p)(as1_void_p)(obase + off),
        (as3_v4i_p)(as3_void_p)(&s_out[off]), 0, 0);
  }
# ifdef HAVE_WAIT_ASYNC
  __builtin_amdgcn_s_wait_asynccnt(0);
# endif
#else
#pragma unroll
  for (int j = 0; j < 12; ++j) {
    const int off = j * 1024 + t * 4;
    *(float4*)(obase + off) = *(const float4*)(&s_out[off]);
  }
  obase[12 * 1024 + t] = s_out[12 * 1024 + t];
#endif
}

// ---------------------------------------------------------------------------
extern "C" void kernel_launch(void* const* d_in, const int* in_sizes, int n_in,
                              void* d_out, int out_size, void* d_ws, size_t ws_size,
                              hipStream_t stream) {
  const float* f[4] = {(const float*)d_in[0], (const float*)d_in[1],
                       (const float*)d_in[2], (const float*)d_in[3]};
  const float* boxes1 = (const float*)d_in[4];
  const float* boxes2 = (const float*)d_in[5];
  const int*   bidx   = (const int*)d_in[6];
  float* out = (float*)d_out;

  const int Hs[4] = {200, 100, 50, 25};
  int B = in_sizes[0] / (CCH * Hs[0] * Hs[0]);
  if (B < 1) B = 1;
  const int N = in_sizes[4] / 4;

  const size_t totalFeat = (size_t)in_sizes[0] + in_sizes[1] +
                           (size_t)in_sizes[2] + in_sizes[3];
  const bool useNHWC = ws_size >= totalFeat * sizeof(float);

  const float* lv[4];
  if (useNHWC) {
    float* nhwc = (float*)d_ws;
    size_t off = 0;
    for (int l = 0; l < 4; ++l) {
      const int HW = Hs[l] * Hs[l];
      dim3 grid((HW + 31) / 32, CCH / 32, B);
      dim3 blk(32, 8, 1);
      nchw_to_nhwc<<<grid, blk, 0, stream>>>(f[l], nhwc + off, CCH, HW);
      lv[l] = nhwc + off;
      off += (size_t)B * HW * CCH;
    }
  } else {
    for (int l = 0; l < 4; ++l) lv[l] = f[l];
  }

  roi_align_pairs<<<N, CCH, 0, stream>>>(lv[0], lv[1], lv[2], lv[3],
                                         boxes1, boxes2, bidx, out,
                                         B, useNHWC ? 1 : 0);
}